// LightGCN_50294067036541
// MI455X (gfx1250) — compile-verified
//
#include <hip/hip_runtime.h>
#include <cstdint>
#include <cstddef>

#define D_DIM 64

typedef __attribute__((ext_vector_type(2))) float v2f;
typedef __attribute__((ext_vector_type(8))) float v8f;

// ---------------------------------------------------------------------------
// Kernel 1: x = concat(emb_user, emb_item); acc = x   (float4 vectorized)
// ---------------------------------------------------------------------------
__global__ void lgcn_init_k(const float* __restrict__ eu, const float* __restrict__ ei,
                            float* __restrict__ x, float* __restrict__ acc,
                            int nu64, int total4) {
    int i = blockIdx.x * blockDim.x + threadIdx.x;
    if (i >= total4) return;
    int base = i * 4;
    float4 v;
    if (base < nu64) v = reinterpret_cast<const float4*>(eu)[i];
    else             v = reinterpret_cast<const float4*>(ei)[i - (nu64 >> 2)];
    reinterpret_cast<float4*>(x)[i]   = v;
    reinterpret_cast<float4*>(acc)[i] = v;
}

// ---------------------------------------------------------------------------
// Kernel 2: edge scatter: x_out[dst] += val * x_in[src]
// One wave per edge; each lane moves a float2 (64 floats / row).
// Uses native global_atomic_add_f32 (L2-resident working set -> L2 atomics).
// ---------------------------------------------------------------------------
__global__ void lgcn_scatter_k(const int* __restrict__ src, const int* __restrict__ dst,
                               const float* __restrict__ val,
                               const float* __restrict__ xin, float* __restrict__ xout,
                               int nE) {
    int t = blockIdx.x * blockDim.x + threadIdx.x;
    int e = t >> 5;
    if (e >= nE) return;
    int lane = t & 31;
    int s = src[e];
    int d = dst[e];
    float v = val[e];
    float2 xv = reinterpret_cast<const float2*>(xin + (size_t)s * D_DIM)[lane];
    float* po = xout + (size_t)d * D_DIM + lane * 2;
    unsafeAtomicAdd(po + 0, v * xv.x);
    unsafeAtomicAdd(po + 1, v * xv.y);
}

// ---------------------------------------------------------------------------
// Kernel 3: acc += x   (float4 vectorized)
// ---------------------------------------------------------------------------
__global__ void lgcn_accadd_k(float* __restrict__ acc, const float* __restrict__ x, int total4) {
    int i = blockIdx.x * blockDim.x + threadIdx.x;
    if (i >= total4) return;
    float4 a = reinterpret_cast<float4*>(acc)[i];
    float4 b = reinterpret_cast<const float4*>(x)[i];
    a.x += b.x; a.y += b.y; a.z += b.z; a.w += b.w;
    reinterpret_cast<float4*>(acc)[i] = a;
}

// ---------------------------------------------------------------------------
// Kernel 4: fused gather + 2x GEMM (V_WMMA_F32_16X16X4_F32) + softmax/sigmoid
//           + row dot -> gamma.
// One wave handles 16 batch rows. Grid is exact (BATCH/16 waves) so EXEC is
// all-ones at every WMMA, as the ISA requires.
//
// A fragment (16x4 F32, 2 VGPRs): lanes 0-15: K=k0,k0+1 ; lanes 16-31: K=k0+2,k0+3
// B fragment (4x16 F32, 2 VGPRs): same K split, lane%16 selects the column.
// C/D (16x16 F32, 8 VGPRs): VGPR r -> row r + 8*(lane>=16), col = lane%16.
// ---------------------------------------------------------------------------
__global__ __launch_bounds__(256) void lgcn_fuse_k(const int* __restrict__ users,
                                                   const int* __restrict__ items,
                                                   const float* __restrict__ acc,
                                                   const float* __restrict__ wu,
                                                   const float* __restrict__ wi,
                                                   float* __restrict__ out,
                                                   int nu) {
    const int lane = threadIdx.x & 31;
    const int hi   = lane >> 4;      // which 16-lane half
    const int l16  = lane & 15;
    const int wid  = blockIdx.x * (blockDim.x >> 5) + (threadIdx.x >> 5);
    const int b0   = wid * 16;       // 16 batch rows per wave

    const v8f vzero = {};

    // ---------------- users @ w_user^T, logits scaled by 1/4 via A ----------
    const int urow = users[b0 + l16];
    const float* pa = acc + (size_t)urow * D_DIM;

    v8f cu[4];
    #pragma unroll
    for (int t = 0; t < 4; ++t) cu[t] = vzero;

    for (int k0 = 0; k0 < D_DIM; k0 += 4) {
        v2f a;
        a.x = pa[k0 + 2 * hi + 0] * 0.25f;
        a.y = pa[k0 + 2 * hi + 1] * 0.25f;
        #pragma unroll
        for (int t = 0; t < 4; ++t) {
            const float* pb = wu + (size_t)(t * 16 + l16) * D_DIM + k0 + 2 * hi;
            v2f b;
            b.x = pb[0];
            b.y = pb[1];
            cu[t] = __builtin_amdgcn_wmma_f32_16x16x4_f32(
                false, a, false, b, (short)0, cu[t], false, false);
        }
    }

    // row-wise softmax over 64 cols (16 lanes x 4 tiles per row)
    #pragma unroll
    for (int r = 0; r < 8; ++r) {
        float m = fmaxf(fmaxf(cu[0][r], cu[1][r]), fmaxf(cu[2][r], cu[3][r]));
        #pragma unroll
        for (int mk = 1; mk < 16; mk <<= 1) m = fmaxf(m, __shfl_xor(m, mk, 32));
        float e0 = __expf(cu[0][r] - m);
        float e1 = __expf(cu[1][r] - m);
        float e2 = __expf(cu[2][r] - m);
        float e3 = __expf(cu[3][r] - m);
        float s = e0 + e1 + e2 + e3;
        #pragma unroll
        for (int mk = 1; mk < 16; mk <<= 1) s += __shfl_xor(s, mk, 32);
        float inv = 1.0f / s;
        cu[0][r] = e0 * inv; cu[1][r] = e1 * inv;
        cu[2][r] = e2 * inv; cu[3][r] = e3 * inv;
    }

    // ---------------- items @ w_item^T ------------------------------------
    const int irow = items[b0 + l16];
    const float* pia = acc + (size_t)(nu + irow) * D_DIM;

    v8f ci[4];
    #pragma unroll
    for (int t = 0; t < 4; ++t) ci[t] = vzero;

    for (int k0 = 0; k0 < D_DIM; k0 += 4) {
        v2f a;
        a.x = pia[k0 + 2 * hi + 0] * 0.25f;
        a.y = pia[k0 + 2 * hi + 1] * 0.25f;
        #pragma unroll
        for (int t = 0; t < 4; ++t) {
            const float* pb = wi + (size_t)(t * 16 + l16) * D_DIM + k0 + 2 * hi;
            v2f b;
            b.x = pb[0];
            b.y = pb[1];
            ci[t] = __builtin_amdgcn_wmma_f32_16x16x4_f32(
                false, a, false, b, (short)0, ci[t], false, false);
        }
    }

    // ---------------- sigmoid + dot product -> gamma ----------------------
    #pragma unroll
    for (int r = 0; r < 8; ++r) {
        float p = 0.0f;
        #pragma unroll
        for (int t = 0; t < 4; ++t) {
            float sg = 1.0f / (1.0f + __expf(-ci[t][r]));
            p += cu[t][r] * sg;
        }
        #pragma unroll
        for (int mk = 1; mk < 16; mk <<= 1) p += __shfl_xor(p, mk, 32);
        if (l16 == 0) out[b0 + r + 8 * hi] = p;
    }
}

// ---------------------------------------------------------------------------
// Host launcher
// ---------------------------------------------------------------------------
extern "C" void kernel_launch(void* const* d_in, const int* in_sizes, int n_in,
                              void* d_out, int out_size, void* d_ws, size_t ws_size,
                              hipStream_t stream) {
    const int*   users = (const int*)d_in[0];
    const int*   items = (const int*)d_in[1];
    const int*   esrc  = (const int*)d_in[2];
    const int*   edst  = (const int*)d_in[3];
    const float* eval  = (const float*)d_in[4];
    const float* eu    = (const float*)d_in[5];
    const float* ei    = (const float*)d_in[6];
    const float* wu    = (const float*)d_in[7];
    const float* wi    = (const float*)d_in[8];
    float* out = (float*)d_out;

    const int batch = in_sizes[0];
    const int nE    = in_sizes[2];
    const int nu    = in_sizes[5] / D_DIM;
    const int ni    = in_sizes[6] / D_DIM;
    const int N     = nu + ni;
    const size_t nElem = (size_t)N * D_DIM;

    float* xa  = (float*)d_ws;
    float* xb  = xa + nElem;
    float* acc = xb + nElem;

    const int threads = 256;
    const int total4  = (int)(nElem / 4);
    const int blk4    = (total4 + threads - 1) / threads;

    // x0 = concat(emb_user, emb_item); acc = x0
    lgcn_init_k<<<blk4, threads, 0, stream>>>(eu, ei, xa, acc, nu * D_DIM, total4);

    // 3 propagation layers
    float* cur = xa;
    float* nxt = xb;
    for (int l = 0; l < 3; ++l) {
        hipMemsetAsync(nxt, 0, nElem * sizeof(float), stream);
        long long tt = (long long)nE * 32;
        int sblocks = (int)((tt + threads - 1) / threads);
        lgcn_scatter_k<<<sblocks, threads, 0, stream>>>(esrc, edst, eval, cur, nxt, nE);
        lgcn_accadd_k<<<blk4, threads, 0, stream>>>(acc, nxt, total4);
        float* tmp = cur; cur = nxt; nxt = tmp;
    }

    // fused gather + WMMA GEMMs + softmax/sigmoid + dot
    int groups = batch / 16;             // one wave per 16 batch rows
    int wavesPerBlock = threads / 32;    // 8
    int fblocks = groups / wavesPerBlock; // exact for batch = 8192
    lgcn_fuse_k<<<fblocks, threads, 0, stream>>>(users, items, acc, wu, wi, out, nu);
}